// CA1Replace_29222957482255
// MI455X (gfx1250) — compile-verified
//
#include <hip/hip_runtime.h>
#include <hip/hip_bf16.h>
#include <stdint.h>

#define CA_W      16384
#define CA_B      32
#define CA_ITERS  64
#define CA_TILE   2048             // cells per wave (32 lanes x 64-bit words)
#define CA_TPR    (CA_W / CA_TILE) // 8 tiles per row
#define CA_GHOST  64               // one 64-bit ghost word per side

// Generic rule application on packed 64-bit words.
// idx bit0 = left neighbour, bit1 = centre, bit2 = right neighbour.
// m[k] is an all-ones/all-zeros mask for lookup[k] (uniform -> SGPRs).
// Each (x & a) | (~x & b) level lowers to v_bfi_b32.
__device__ __forceinline__ uint64_t rule_apply(uint64_t L, uint64_t C, uint64_t R,
                                               const uint64_t* __restrict__ m) {
  uint64_t p0 = (L & m[1]) | (~L & m[0]);   // C=0,R=0
  uint64_t p1 = (L & m[3]) | (~L & m[2]);   // C=1,R=0
  uint64_t p2 = (L & m[5]) | (~L & m[4]);   // C=0,R=1
  uint64_t p3 = (L & m[7]) | (~L & m[6]);   // C=1,R=1
  uint64_t q0 = (C & p1) | (~C & p0);
  uint64_t q1 = (C & p3) | (~C & p2);
  return (R & q1) | (~R & q0);
}

// Unpack one 64-bit word to 64 floats via the LDS nibble->float4 table:
// 16x (v_bfe + ds_load_b128) feeding 16x global_store_b128.
__device__ __forceinline__ void store_word_t(float4* __restrict__ o, uint64_t w,
                                             const float4* __restrict__ tab) {
  uint32_t lo = (uint32_t)w;
  uint32_t hi = (uint32_t)(w >> 32);
#pragma unroll
  for (int q = 0; q < 8; ++q) {
    o[q]     = tab[(lo >> (4 * q)) & 0xFu];
    o[q + 8] = tab[(hi >> (4 * q)) & 0xFu];
  }
}

__global__ __launch_bounds__(32) void CA1Replace_rule110_kernel(
    const float* __restrict__ x, const int* __restrict__ lookup,
    float* __restrict__ out) {
  // Tile + two ghost words of input floats staged through LDS (8704 B),
  // plus a 16-entry nibble->float4 unpack table (256 B).
  __shared__ __align__(16) float stage[CA_TILE + 2 * CA_GHOST];
  __shared__ __align__(16) float4 tab[16];

  const int lane = threadIdx.x;            // 0..31, wave32
  const int b    = blockIdx.x / CA_TPR;    // row
  const int t    = blockIdx.x % CA_TPR;    // tile in row
  const int tile_start = t * CA_TILE;

  // Rule masks from runtime lookup table (uniform -> scalar regs).
  uint64_t m[8];
#pragma unroll
  for (int k = 0; k < 8; ++k) m[k] = lookup[k] ? ~0ull : 0ull;

  // Build the unpack table (same-wave LDS ordering; no barrier needed).
  if (lane < 16) {
    tab[lane] = make_float4((float)(lane & 1), (float)((lane >> 1) & 1),
                            (float)((lane >> 2) & 1), (float)((lane >> 3) & 1));
  }

  // ---- async copy of input tile (+halos) into LDS (CDNA5 ASYNCcnt path) ----
  {
    uint32_t ldsBase = (uint32_t)(uintptr_t)(&stage[0]);
    long long gbase = ((long long)b * CA_W + tile_start - CA_GHOST) * 4ll;
    const long long gmax = (long long)CA_B * CA_W * 4ll - 16ll;
    unsigned long long xs = (unsigned long long)(uintptr_t)x;
#pragma unroll
    for (int c = 0; c < 17; ++c) {          // 17 * 32 lanes * 16B = 8704 B
      int e = c * 32 + lane;                // 16-byte chunk index
      uint32_t ldsOff = ldsBase + (uint32_t)e * 16u;
      long long g = gbase + (long long)e * 16ll;
      if (g < 0) g = 0;                     // clamp: corner tiles only; ghost
      if (g > gmax) g = gmax;               // bits are forced to 0 below
      uint32_t gOff = (uint32_t)g;
      asm volatile("global_load_async_to_lds_b128 %0, %1, %2 offset:0"
                   :: "v"(ldsOff), "v"(gOff), "s"(xs) : "memory");
    }
    asm volatile("s_wait_asynccnt 0" ::: "memory");
  }

  // ---- threshold to packed bits -------------------------------------------
  const float4* sv = (const float4*)stage;
  uint64_t w = 0;
  {
    int base4 = (CA_GHOST + lane * 64) >> 2;
#pragma unroll
    for (int q = 0; q < 16; ++q) {
      float4 v = sv[base4 + q];
      uint64_t bits = 0;
      bits |= (v.x >= 0.5f) ? 1ull : 0ull;
      bits |= (v.y >= 0.5f) ? 2ull : 0ull;
      bits |= (v.z >= 0.5f) ? 4ull : 0ull;
      bits |= (v.w >= 0.5f) ? 8ull : 0ull;
      w |= bits << (q * 4);
    }
  }
  // Ghost word: lane 0 holds the left ghost, lane 31 the right ghost.
  uint64_t ghost = 0;
  {
    int gbase4 = (lane == 31) ? ((CA_GHOST + CA_TILE) >> 2) : 0;
#pragma unroll
    for (int q = 0; q < 16; ++q) {
      float4 v = sv[gbase4 + q];
      uint64_t bits = 0;
      bits |= (v.x >= 0.5f) ? 1ull : 0ull;
      bits |= (v.y >= 0.5f) ? 2ull : 0ull;
      bits |= (v.z >= 0.5f) ? 4ull : 0ull;
      bits |= (v.w >= 0.5f) ? 8ull : 0ull;
      ghost |= bits << (q * 4);
    }
  }
  const bool leftEdgeLane  = (lane == 0)  && (t == 0);
  const bool rightEdgeLane = (lane == 31) && (t == CA_TPR - 1);
  if (leftEdgeLane || rightEdgeLane) ghost = 0;  // reference zero-pads each step

  // Hoisted output pointer: advance one row (CA_W floats) per iteration.
  const int cellBase = tile_start + lane * 64;
  float4* op = (float4*)(out + (size_t)b * (CA_ITERS + 1) * CA_W + cellBase);
  store_word_t(op, w, tab);                      // iteration 0 = thresholded input

  // ---- 64 iterations, wave-synchronous (no barriers) ----------------------
#pragma unroll 1
  for (int it = 1; it <= CA_ITERS; ++it) {
    // Halo carries via ballot masks (no ds_bpermute in the dependence chain).
    uint32_t msb = (uint32_t)(w >> 63) & 1u;
    uint32_t lsb = (uint32_t)w & 1u;
    uint32_t msbMask = (uint32_t)__ballot(msb);
    uint32_t lsbMask = (uint32_t)__ballot(lsb);
    uint32_t cl = ((msbMask << 1) >> lane) & 1u; // msb of lane-1's word
    uint32_t cr = ((lsbMask >> 1) >> lane) & 1u; // lsb of lane+1's word
    if (lane == 0)  cl = (uint32_t)(ghost >> 63) & 1u;
    if (lane == 31) cr = (uint32_t)(ghost) & 1u;

    uint64_t L = (w << 1) | (uint64_t)cl;
    uint64_t R = (w >> 1) | ((uint64_t)cr << 63);
    uint64_t nw = rule_apply(L, w, R, m);

    // Evolve the ghost word (exact on its inner side; outer-edge errors never
    // reach the main words within 64 iterations).
    uint64_t gL = (lane == 31) ? ((ghost << 1) | (uint64_t)msb)
                               : (ghost << 1);
    uint64_t gR = (lane == 31) ? (ghost >> 1)
                               : ((ghost >> 1) | ((uint64_t)lsb << 63));
    uint64_t ng = rule_apply(gL, ghost, gR, m);
    if (leftEdgeLane || rightEdgeLane) ng = 0;   // true row edges stay zero

    w = nw;
    ghost = ng;
    op += CA_W / 4;
    store_word_t(op, w, tab);
  }
}

extern "C" void kernel_launch(void* const* d_in, const int* in_sizes, int n_in,
                              void* d_out, int out_size, void* d_ws, size_t ws_size,
                              hipStream_t stream) {
  (void)in_sizes; (void)n_in; (void)out_size; (void)d_ws; (void)ws_size;
  const float* x      = (const float*)d_in[0];
  const int*   lookup = (const int*)d_in[1];
  float*       out    = (float*)d_out;
  dim3 grid(CA_B * CA_TPR);  // 256 single-wave workgroups
  dim3 block(32);
  hipLaunchKernelGGL(CA1Replace_rule110_kernel, grid, block, 0, stream,
                     x, lookup, out);
}